// MinGRUCell_35287451304712
// MI455X (gfx1250) — compile-verified
//
#include <hip/hip_runtime.h>
#include <math.h>

// Problem constants (match reference)
#define BB      8
#define SS      8192
#define IN_DIM  256
#define HH      256
#define LCHUNK  64                    // sequence rows handled per workgroup (chunk)
#define NCHUNK  (SS / LCHUNK)         // 128 chunks per batch
#define NBLK    (BB * NCHUNK)         // 1024 workgroups

// LDS layout (floats), padded strides for bank-conflict-free access
#define XS_STRIDE 260                 // x tile row stride (260 mod 64 banks = 4)
#define CV_STRIDE 514                 // lc/lv pair-tile row stride (512 data + 2 pad)
#define XS_FLOATS (LCHUNK * XS_STRIDE)            // 16640
#define CV_FLOATS (LCHUNK * CV_STRIDE)            // 32896
#define LDS_BYTES ((XS_FLOATS + CV_FLOATS) * 4)   // 198144 B < 320 KB WGP LDS

typedef __attribute__((ext_vector_type(2))) float v2f;
typedef __attribute__((ext_vector_type(8))) float v8f;
typedef __attribute__((ext_vector_type(4))) int   v4i;

// Address-space-qualified 16B vector pointers for the async-LDS builtin
typedef __attribute__((address_space(1))) v4i gv4i;   // global
typedef __attribute__((address_space(3))) v4i lv4i;   // LDS

#if __has_builtin(__builtin_amdgcn_global_load_async_to_lds_b128)
#define HAS_ASYNC_LDS 1
#endif

__device__ __forceinline__ void wait_async_zero() {
#if __has_builtin(__builtin_amdgcn_s_wait_asynccnt)
    __builtin_amdgcn_s_wait_asynccnt(0);
#else
    asm volatile("s_wait_asynccnt 0x0" ::: "memory");
#endif
}

// Fast stable softplus: two hardware transcendentals, no libcalls.
// Absolute error ~2^-24 in a log-magnitude result -- fp32-level agreement.
__device__ __forceinline__ float softplus_f(float x) {
    return fmaxf(x, 0.0f) + __logf(1.0f + __expf(-fabsf(x)));
}

// logaddexp without the -inf guard: in this problem lv is always finite, so
// the -inf-initial state resolves via exp(-inf)=0 with no branch.
__device__ __forceinline__ float logaddexp_f(float a, float b) {
    float m = fmaxf(a, b);
    return m + __logf(1.0f + __expf(fminf(a, b) - m));
}

// log g(x): branchless select of log(x+0.5) (x>=0) vs log(sigmoid(x)) (x<0).
__device__ __forceinline__ float log_g_f(float x) {
    float pos = __logf(x + 0.5f);                   // NaN for unused side is fine
    float neg = x - __logf(1.0f + __expf(x));       // -softplus(-x), x<0 branch
    return (x >= 0.0f) ? pos : neg;
}

// Stage the 64x256 f32 x tile into LDS (padded stride) with async copies:
// each of 256 threads moves 16 B per sweep, 16 sweeps total (ASYNCcnt path).
__device__ __forceinline__ void stage_x_tile(const float* __restrict__ x,
                                             int m0, float* __restrict__ xs)
{
    const int tid = threadIdx.x;
    const float* __restrict__ src = x + (size_t)m0 * IN_DIM;
    #pragma unroll
    for (int sweep = 0; sweep < (LCHUNK * IN_DIM) / (256 * 4); ++sweep) {
        const int e   = (sweep * 256 + tid) * 4;     // float index in tile
        const int row = e >> 8;                      // /IN_DIM
        const int col = e & (IN_DIM - 1);
        float* dst = xs + row * XS_STRIDE + col;
#ifdef HAS_ASYNC_LDS
        __builtin_amdgcn_global_load_async_to_lds_b128(
            (gv4i*)(src + e), (lv4i*)dst, 0, 0);
#else
        *(float4*)dst = *(const float4*)(src + e);
#endif
    }
#ifdef HAS_ASYNC_LDS
    wait_async_zero();
#endif
}

// GEMM + pointwise for global rows [m0, m0+64): waves own N-column pairs
// (wave w -> gate cols [w*32, w*32+32) and matching hidden cols) for ALL rows,
// so each W element is read exactly once per WG and one A fragment feeds
// 8 WMMAs (4 M-tiles x {gate,hidden}). Results land in the LDS {lc,lv} tile,
// interleaved as float2 per (row, h) for single-b64 access in the scan phase.
__device__ __forceinline__ void compute_tile_lds(
    const float* __restrict__ W, const float* __restrict__ bias,
    const float* __restrict__ xs /* LDS x tile */, float* __restrict__ cv /* LDS {lc,lv} */)
{
    const int tid  = threadIdx.x;
    const int lane = tid & 31;
    const int wave = tid >> 5;      // 0..7
    const int hl   = lane >> 4;     // lane-half: selects K pair (A/B) and M+8 (C)
    const int l15  = lane & 15;

    for (int p = 0; p < 2; ++p) {                    // two column pairs per wave
        const int ng = (wave * 2 + p) * 16;          // gate column base (0..255)
        const float* __restrict__ wg_row = W + (size_t)(ng + l15) * IN_DIM;
        const float* __restrict__ wh_row = W + (size_t)(HH + ng + l15) * IN_DIM;

        v8f cg0 = {}, cg1 = {}, cg2 = {}, cg3 = {};  // gate accs, M-tiles 0..3
        v8f ch0 = {}, ch1 = {}, ch2 = {}, ch3 = {};  // hidden accs
        for (int kk = 0; kk < IN_DIM / 4; ++kk) {
            const int k0 = kk * 4 + hl * 2;
            v2f bg = *(const v2f*)(wg_row + k0);     // B[k][n] = W[n][k]
            v2f bh = *(const v2f*)(wh_row + k0);
            v2f a0 = *(const v2f*)(xs + (0 * 16 + l15) * XS_STRIDE + k0);
            v2f a1 = *(const v2f*)(xs + (1 * 16 + l15) * XS_STRIDE + k0);
            v2f a2 = *(const v2f*)(xs + (2 * 16 + l15) * XS_STRIDE + k0);
            v2f a3 = *(const v2f*)(xs + (3 * 16 + l15) * XS_STRIDE + k0);
            cg0 = __builtin_amdgcn_wmma_f32_16x16x4_f32(false, a0, false, bg, (short)0, cg0, false, false);
            ch0 = __builtin_amdgcn_wmma_f32_16x16x4_f32(false, a0, false, bh, (short)0, ch0, false, false);
            cg1 = __builtin_amdgcn_wmma_f32_16x16x4_f32(false, a1, false, bg, (short)0, cg1, false, false);
            ch1 = __builtin_amdgcn_wmma_f32_16x16x4_f32(false, a1, false, bh, (short)0, ch1, false, false);
            cg2 = __builtin_amdgcn_wmma_f32_16x16x4_f32(false, a2, false, bg, (short)0, cg2, false, false);
            ch2 = __builtin_amdgcn_wmma_f32_16x16x4_f32(false, a2, false, bh, (short)0, ch2, false, false);
            cg3 = __builtin_amdgcn_wmma_f32_16x16x4_f32(false, a3, false, bg, (short)0, cg3, false, false);
            ch3 = __builtin_amdgcn_wmma_f32_16x16x4_f32(false, a3, false, bh, (short)0, ch3, false, false);
        }

        const int   n    = ng + l15;                 // hidden-channel index (0..255)
        const float bgte = bias[n];
        const float bhid = bias[HH + n];
        v8f cgs[4] = {cg0, cg1, cg2, cg3};
        v8f chs[4] = {ch0, ch1, ch2, ch3};
        #pragma unroll
        for (int t = 0; t < 4; ++t) {
            #pragma unroll
            for (int r = 0; r < 8; ++r) {
                const int ml = t * 16 + hl * 8 + r;  // local row within the chunk
                float gate   = cgs[t][r] + bgte;
                float hidden = chs[t][r] + bhid;
                v2f pr;
                pr.x = -softplus_f(gate);                    // lc = log(1 - z)
                pr.y = -softplus_f(-gate) + log_g_f(hidden); // lv = log z + log g
                *(v2f*)(cv + ml * CV_STRIDE + 2 * n) = pr;   // one ds_store_b64
            }
        }
    }
}

// Phase 1: per-chunk summaries (la = sum lc, lb = local scan from -inf)
__global__ void __launch_bounds__(256)
mingru_gemm_summary(const float* __restrict__ x, const float* __restrict__ W,
                    const float* __restrict__ bias,
                    float* __restrict__ sums /* {la,lb} pairs */)
{
    extern __shared__ float lds[];
    float* xs = lds;                 // x tile
    float* cv = lds + XS_FLOATS;     // {lc,lv} tile
    const int g  = blockIdx.x;       // chunk id: b = g / NCHUNK
    const int m0 = g * LCHUNK;
    stage_x_tile(x, m0, xs);
    __syncthreads();
    compute_tile_lds(W, bias, xs, cv);
    __syncthreads();

    const int h = threadIdx.x;       // one thread per hidden channel
    float la = 0.0f, lb = -INFINITY;
    for (int i = 0; i < LCHUNK; ++i) {
        v2f pr = *(const v2f*)(cv + i * CV_STRIDE + 2 * h);  // one ds_load_b64
        la += pr.x;
        lb  = logaddexp_f(pr.x + lb, pr.y);
    }
    v2f s; s.x = la; s.y = lb;
    *(v2f*)(sums + ((size_t)g * HH + h) * 2) = s;
}

// Phase 2: sequential scan over the 128 chunk summaries per (b, h)
__global__ void __launch_bounds__(256)
mingru_chunk_scan(const float* __restrict__ h0, const float* __restrict__ sums,
                  float* __restrict__ lh_start)
{
    const int b = blockIdx.x;        // 0..7
    const int h = threadIdx.x;       // 0..255
    float lh = __logf(h0[b * HH + h]);
    for (int c = 0; c < NCHUNK; ++c) {
        const int g = b * NCHUNK + c;
        lh_start[g * HH + h] = lh;   // state entering chunk c
        v2f s = *(const v2f*)(sums + ((size_t)g * HH + h) * 2);
        lh = logaddexp_f(s.x + lh, s.y);
    }
}

// Phase 3: recompute GEMM tile, apply recurrence from known start state, emit exp(lh)
__global__ void __launch_bounds__(256)
mingru_gemm_apply(const float* __restrict__ x, const float* __restrict__ W,
                  const float* __restrict__ bias,
                  const float* __restrict__ lh_start,
                  float* __restrict__ out, float* __restrict__ hnext)
{
    extern __shared__ float lds[];
    float* xs = lds;
    float* cv = lds + XS_FLOATS;
    const int g  = blockIdx.x;
    const int m0 = g * LCHUNK;
    stage_x_tile(x, m0, xs);
    __syncthreads();
    compute_tile_lds(W, bias, xs, cv);
    __syncthreads();

    const int h = threadIdx.x;
    float lh = lh_start[g * HH + h];
    float* __restrict__ orow = out + (size_t)m0 * HH;
    for (int i = 0; i < LCHUNK; ++i) {
        v2f pr = *(const v2f*)(cv + i * CV_STRIDE + 2 * h);
        lh = logaddexp_f(pr.x + lh, pr.y);
        __builtin_nontemporal_store(__expf(lh), &orow[(size_t)i * HH + h]); // streamed once
    }
    if ((g % NCHUNK) == NCHUNK - 1) {            // last chunk of this batch
        hnext[(g / NCHUNK) * HH + h] = __expf(lh);
    }
}

extern "C" void kernel_launch(void* const* d_in, const int* in_sizes, int n_in,
                              void* d_out, int out_size, void* d_ws, size_t ws_size,
                              hipStream_t stream) {
    const float* x    = (const float*)d_in[0];   // (B, S, IN)
    const float* h0   = (const float*)d_in[1];   // (B, 1, H)
    const float* W    = (const float*)d_in[2];   // (2H, IN)
    const float* bias = (const float*)d_in[3];   // (2H,)

    float* out   = (float*)d_out;                            // (B, S, H)
    float* hnext = out + (size_t)BB * SS * HH;               // (B, 1, H)

    // Workspace: interleaved {la,lb} chunk summaries + chunk start states (3 MB)
    float* sums     = (float*)d_ws;                          // NBLK*HH*2 floats
    float* lh_start = sums + (size_t)NBLK * HH * 2;          // NBLK*HH floats

    mingru_gemm_summary<<<NBLK, 256, LDS_BYTES, stream>>>(x, W, bias, sums);
    mingru_chunk_scan  <<<BB,   256, 0,         stream>>>(h0, sums, lh_start);
    mingru_gemm_apply  <<<NBLK, 256, LDS_BYTES, stream>>>(x, W, bias, lh_start, out, hnext);
}